// CongestionWrapperEncoder_37915971289819
// MI455X (gfx1250) — compile-verified
//
#include <hip/hip_runtime.h>

typedef __attribute__((ext_vector_type(2))) float v2f;
typedef __attribute__((ext_vector_type(8))) float v8f;

#define G    32      // B*D graphs
#define NN   2000    // nodes per graph
#define EMBD 32      // embedding dim
#define HH   4       // heads
#define HC   128     // H*C
#define EE   2000    // raw edges
#define ET   4000    // edges + self loops
#define WC   144     // augmented weight cols: 128 + 4(src) + 4(dst) + 8 pad
#define NTILES 125   // 2000 / 16

// ---------- helpers ----------
__device__ __forceinline__ int fenc(float f) {           // monotone float -> int key
  int i = __float_as_int(f);
  return i >= 0 ? i : (i ^ 0x7fffffff);
}
__device__ __forceinline__ float fdec(int i) {
  return __int_as_float(i >= 0 ? i : (i ^ 0x7fffffff));
}
__device__ __forceinline__ float lrelu(float v) { return v > 0.f ? v : 0.2f * v; }

// ---------- 1) augmented weight: [32 x 144] = [lin_w | W^T att_src | W^T att_dst | 0] ----------
__global__ void k_waug(const float* __restrict__ lin, const float* __restrict__ asv,
                       const float* __restrict__ adv, float* __restrict__ waug) {
  int d = blockIdx.x;            // 0..31
  int col = threadIdx.x;         // 0..159
  if (col >= WC) return;
  float v = 0.f;
  if (col < 128) {
    v = lin[d * HC + col];
  } else if (col < 132) {
    int h = col - 128; float a = 0.f;
    for (int c = 0; c < 32; ++c) a += lin[d * HC + h * 32 + c] * asv[h * 32 + c];
    v = a;
  } else if (col < 136) {
    int h = col - 132; float a = 0.f;
    for (int c = 0; c < 32; ++c) a += lin[d * HC + h * 32 + c] * adv[h * 32 + c];
    v = a;
  }
  waug[d * WC + col] = v;
}

// ---------- 2) fused embedding-gather + GEMM via V_WMMA_F32_16X16X4_F32 ----------
// one wave (32 lanes) per (graph, 16-node tile); 9 column tiles x 8 K-steps
__global__ void __launch_bounds__(32) k_gemm(const int* __restrict__ x,
                                             const float* __restrict__ emb,
                                             const float* __restrict__ waug,
                                             float* __restrict__ hl,
                                             float* __restrict__ asrc,
                                             float* __restrict__ adst) {
  int g = blockIdx.x / NTILES;
  int t = blockIdx.x % NTILES;
  int lane = threadIdx.x;
  int row = lane & 15;           // A: M index / B,C,D: N index
  int kb  = (lane >> 4) * 2;     // A/B: K sub-offset within 4-wide slice
  int node = t * 16 + row;
  int xi = x[g * NN + node];
  const float* erow = emb + (size_t)xi * EMBD;

  // preload all A fragments (EMB=32 -> 8 K-slices of 4)
  v2f a[8];
#pragma unroll
  for (int k = 0; k < 8; ++k) {
    a[k].x = erow[k * 4 + kb];
    a[k].y = erow[k * 4 + kb + 1];
  }

  int mbase = (lane >> 4) * 8;   // C/D: rows held by this lane = mbase..mbase+7
  for (int j = 0; j < 9; ++j) {
    v8f acc = {};
#pragma unroll
    for (int k = 0; k < 8; ++k) {
      int kr = k * 4 + kb;
      v2f b;
      b.x = waug[kr * WC + j * 16 + row];
      b.y = waug[(kr + 1) * WC + j * 16 + row];
      acc = __builtin_amdgcn_wmma_f32_16x16x4_f32(false, a[k], false, b,
                                                  (short)0, acc, false, false);
    }
    if (j < 8) {
#pragma unroll
      for (int i = 0; i < 8; ++i)
        hl[(size_t)(g * NN + t * 16 + mbase + i) * HC + j * 16 + row] = acc[i];
    } else {
      if (row < 4) {
#pragma unroll
        for (int i = 0; i < 8; ++i)
          asrc[(size_t)(g * NN + t * 16 + mbase + i) * HH + row] = acc[i];
      } else if (row < 8) {
#pragma unroll
        for (int i = 0; i < 8; ++i)
          adst[(size_t)(g * NN + t * 16 + mbase + i) * HH + (row - 4)] = acc[i];
      }
    }
  }
}

// ---------- 3) init ----------
__global__ void k_init_out(float* __restrict__ out, const float* __restrict__ bias, int n) {
  int i = blockIdx.x * blockDim.x + threadIdx.x;
  if (i < n) out[i] = bias[i & (HC - 1)];
}
__global__ void k_init_ms(int* __restrict__ mI, float* __restrict__ sS, int n) {
  int i = blockIdx.x * blockDim.x + threadIdx.x;
  if (i < n) { mI[i] = (int)0x80000000; sS[i] = 0.f; }
}

// ---------- 4) segment max over dst ----------
__global__ void k_max(const int* __restrict__ adj, const float* __restrict__ asrc,
                      const float* __restrict__ adst, int* __restrict__ mI) {
  int idx = blockIdx.x * blockDim.x + threadIdx.x;
  if (idx >= G * ET * HH) return;
  int h = idx & 3; int rest = idx >> 2;
  int e = rest % ET; int g = rest / ET;
  int sN, dN;
  if (e < EE) { sN = adj[e]; dN = adj[EE + e]; } else { sN = dN = e - EE; }
  float lg = lrelu(asrc[(g * NN + sN) * HH + h] + adst[(g * NN + dN) * HH + h]);
  atomicMax(&mI[(g * NN + dN) * HH + h], fenc(lg));
}

// ---------- 5) segment exp-sum ----------
__global__ void k_sum(const int* __restrict__ adj, const float* __restrict__ asrc,
                      const float* __restrict__ adst, const int* __restrict__ mI,
                      float* __restrict__ sS) {
  int idx = blockIdx.x * blockDim.x + threadIdx.x;
  if (idx >= G * ET * HH) return;
  int h = idx & 3; int rest = idx >> 2;
  int e = rest % ET; int g = rest / ET;
  int sN, dN;
  if (e < EE) { sN = adj[e]; dN = adj[EE + e]; } else { sN = dN = e - EE; }
  int di = (g * NN + dN) * HH + h;
  float lg = lrelu(asrc[(g * NN + sN) * HH + h] + adst[di]);
  float ex = expf(lg - fdec(mI[di]));
  atomicAdd(&sS[di], ex);
}

// ---------- 6) alpha-weighted message scatter ----------
__global__ void __launch_bounds__(HC) k_scatter(const int* __restrict__ adj,
                                                const float* __restrict__ asrc,
                                                const float* __restrict__ adst,
                                                const int* __restrict__ mI,
                                                const float* __restrict__ sS,
                                                const float* __restrict__ hl,
                                                float* __restrict__ out) {
  int e = blockIdx.x % ET;
  int g = blockIdx.x / ET;
  int sN, dN;
  if (e < EE) { sN = adj[e]; dN = adj[EE + e]; } else { sN = dN = e - EE; }
  __shared__ float alpha[HH];
  if (threadIdx.x < HH) {
    int h = threadIdx.x;
    int di = (g * NN + dN) * HH + h;
    float lg = lrelu(asrc[(g * NN + sN) * HH + h] + adst[di]);
    float ex = expf(lg - fdec(mI[di]));
    alpha[h] = ex / sS[di];
  }
  __syncthreads();
  int hc = threadIdx.x;
  float v = alpha[hc >> 5] * hl[(size_t)(g * NN + sN) * HC + hc];
  atomicAdd(&out[(size_t)(g * NN + dN) * HC + hc], v);
}

extern "C" void kernel_launch(void* const* d_in, const int* in_sizes, int n_in,
                              void* d_out, int out_size, void* d_ws, size_t ws_size,
                              hipStream_t stream) {
  const int*   x    = (const int*)d_in[0];     // [B,D,N] int32
  const int*   adj  = (const int*)d_in[1];     // [2,E]   int32
  const float* emb  = (const float*)d_in[2];   // [N,32]
  const float* lin  = (const float*)d_in[3];   // [32,128]
  const float* asv  = (const float*)d_in[4];   // [4,32]
  const float* adv  = (const float*)d_in[5];   // [4,32]
  const float* bias = (const float*)d_in[6];   // [128]
  float* out = (float*)d_out;                  // G*N*128 f32

  // workspace layout (floats)
  float* hl   = (float*)d_ws;                  // G*N*128 = 8,192,000
  float* asrc = hl + (size_t)G * NN * HC;      // G*N*4
  float* adst = asrc + (size_t)G * NN * HH;    // G*N*4
  int*   mI   = (int*)(adst + (size_t)G * NN * HH);
  float* sS   = (float*)(mI + (size_t)G * NN * HH);
  float* waug = sS + (size_t)G * NN * HH;      // 32*144

  // 1) augmented weights
  k_waug<<<EMBD, 160, 0, stream>>>(lin, asv, adv, waug);

  // 2) WMMA GEMM: hl + attention logits
  k_gemm<<<G * NTILES, 32, 0, stream>>>(x, emb, waug, hl, asrc, adst);

  // 3) init out (bias) and segment max/sum buffers
  {
    int n = G * NN * HC;
    k_init_out<<<(n + 255) / 256, 256, 0, stream>>>(out, bias, n);
  }
  {
    int n = G * NN * HH;
    k_init_ms<<<(n + 255) / 256, 256, 0, stream>>>(mI, sS, n);
  }

  // 4) segment max
  {
    int n = G * ET * HH;
    k_max<<<(n + 255) / 256, 256, 0, stream>>>(adj, asrc, adst, mI);
  }

  // 5) segment exp-sum
  {
    int n = G * ET * HH;
    k_sum<<<(n + 255) / 256, 256, 0, stream>>>(adj, asrc, adst, mI, sS);
  }

  // 6) scatter alpha * hl[src] into out[dst]
  k_scatter<<<G * ET, HC, 0, stream>>>(adj, asrc, adst, mI, sS, hl, out);
}